// GraphEncoder_67826123538492
// MI455X (gfx1250) — compile-verified
//
#include <hip/hip_runtime.h>

typedef __attribute__((ext_vector_type(2))) float v2f;
typedef __attribute__((ext_vector_type(8))) float v8f;

#define HID 128
#define LAT 64

__device__ __forceinline__ void atomAddF(float* p, float v) {
  __hip_atomic_fetch_add(p, v, __ATOMIC_RELAXED, __HIP_MEMORY_SCOPE_AGENT);
}

// ---------------- degree / normalization ----------------

__global__ void k_init_deg(float* __restrict__ deg, int N) {
  int i = blockIdx.x * blockDim.x + threadIdx.x;
  if (i < N) deg[i] = 1.0f;  // self-loop contributes 1
}

__global__ void k_zero_g(float* __restrict__ g) {
  if (threadIdx.x < HID) g[threadIdx.x] = 0.0f;
}

__global__ void k_deg_accum(const int* __restrict__ dst, float* __restrict__ deg, int E) {
  int e = blockIdx.x * blockDim.x + threadIdx.x;
  if (e < E) atomAddF(&deg[dst[e]], 1.0f);
}

__global__ void k_rsqrt_inplace(float* __restrict__ deg, int N) {
  int i = blockIdx.x * blockDim.x + threadIdx.x;
  if (i < N) deg[i] = rsqrtf(deg[i]);  // deg >= 1 always
}

// ---------------- dense GEMM via f32 WMMA (16x16x4) ----------------
// H[N,128] = X[N,K] @ W[K,128].  N multiple of 16 (100000 = 6250*16), K multiple of 4.
// block = 256 threads = 8 waves; wave w computes the 16x16 tile at cols w*16.
__global__ void k_gemm_wmma_f32(const float* __restrict__ X,
                                const float* __restrict__ W,
                                float* __restrict__ H,
                                int N, int K) {
  const int Cout = HID;
  const int r0   = blockIdx.x * 16;
  const int wave = threadIdx.x >> 5;   // 0..7 -> column tile
  const int lane = threadIdx.x & 31;
  const int c0   = wave * 16;
  const int m    = lane & 15;          // row (A) / col (B,D) within tile
  const int half = lane >> 4;          // K sub-pair select

  v8f c = {};
  const float* xrow = X + (size_t)(r0 + m) * K;

  for (int kk = 0; kk < K; kk += 4) {
    const int ka = kk + half * 2;
    v2f a, b;
    // A 16x4 f32: VGPR0 = K=ka, VGPR1 = K=ka+1 (ISA 7.12.2)
    a.x = xrow[ka];
    a.y = xrow[ka + 1];
    // B 4x16 f32: row K=ka striped over lanes (N = lane&15), mirrored halves
    b.x = W[(size_t)ka * Cout + c0 + m];
    b.y = W[(size_t)(ka + 1) * Cout + c0 + m];
    c = __builtin_amdgcn_wmma_f32_16x16x4_f32(false, a, false, b, (short)0, c,
                                              false, false);
  }

  // D 16x16 f32: VGPR r -> M = r + 8*half, N = lane&15
  float* hp = H + (size_t)r0 * Cout + c0 + m;
  const int mbase = half * 8;
#pragma unroll
  for (int r = 0; r < 8; ++r) {
    hp[(size_t)(mbase + r) * Cout] = c[r];
  }
}

// ---------------- self-loop init: acc[i] = h[i] * dinv[i]^2 ----------------
__global__ void k_selfloop_init(const float* __restrict__ H,
                                const float* __restrict__ dinv,
                                float* __restrict__ acc, int N) {
  size_t idx = blockIdx.x * (size_t)blockDim.x + threadIdx.x;  // over N*32 float4s
  if (idx >= (size_t)N * (HID / 4)) return;
  int i = (int)(idx >> 5);
  float d = dinv[i];
  d = d * d;
  float4 v = ((const float4*)H)[idx];
  v.x *= d; v.y *= d; v.z *= d; v.w *= d;
  ((float4*)acc)[idx] = v;
}

// ---------------- edge scatter: one wave per edge, 32 lanes x float4 ----------------
__global__ void k_scatter(const float* __restrict__ H,
                          const int* __restrict__ src,
                          const int* __restrict__ dst,
                          const float* __restrict__ dinv,
                          float* __restrict__ acc, int E) {
  int wave   = (int)((blockIdx.x * (size_t)blockDim.x + threadIdx.x) >> 5);
  int lane   = threadIdx.x & 31;
  int nwaves = (int)((gridDim.x * (size_t)blockDim.x) >> 5);
  for (int e = wave; e < E; e += nwaves) {
    int s = src[e];
    int d = dst[e];
    float nrm = dinv[s] * dinv[d];
    float4 v = ((const float4*)(H + (size_t)s * HID))[lane];
    float* ap = acc + (size_t)d * HID + lane * 4;
    atomAddF(ap + 0, v.x * nrm);
    atomAddF(ap + 1, v.y * nrm);
    atomAddF(ap + 2, v.z * nrm);
    atomAddF(ap + 3, v.w * nrm);
  }
}

// ---------------- bias + ReLU (in place) ----------------
__global__ void k_bias_relu(float* __restrict__ acc, const float* __restrict__ b, int N) {
  size_t idx = blockIdx.x * (size_t)blockDim.x + threadIdx.x;  // over N*32 float4s
  if (idx >= (size_t)N * (HID / 4)) return;
  int f4 = (int)(idx & 31);
  float4 bb = ((const float4*)b)[f4];
  float4 v = ((float4*)acc)[idx];
  v.x = fmaxf(v.x + bb.x, 0.0f);
  v.y = fmaxf(v.y + bb.y, 0.0f);
  v.z = fmaxf(v.z + bb.z, 0.0f);
  v.w = fmaxf(v.w + bb.w, 0.0f);
  ((float4*)acc)[idx] = v;
}

// ---------------- mean pool: g[f] = sum_i A[i][f] (scaled later) ----------------
__global__ void k_pool(const float* __restrict__ A, float* __restrict__ g, int N) {
  int f = threadIdx.x;  // 128 threads
  float s = 0.0f;
  for (int i = blockIdx.x; i < N; i += gridDim.x) s += A[(size_t)i * HID + f];
  atomAddF(&g[f], s);
}

// ---------------- heads: mu = g@Wmu + bmu ; logvar = g@Wlv + blv ----------------
__global__ void k_heads(const float* __restrict__ g,
                        const float* __restrict__ Wmu, const float* __restrict__ bmu,
                        const float* __restrict__ Wlv, const float* __restrict__ blv,
                        float* __restrict__ out, float invN) {
  int j = threadIdx.x;  // 0..127
  const float* Wm = (j < LAT) ? Wmu : Wlv;
  const float* bb = (j < LAT) ? bmu : blv;
  int jj = j & (LAT - 1);
  float acc = bb[jj];
  for (int f = 0; f < HID; ++f) acc += (g[f] * invN) * Wm[f * LAT + jj];
  out[j] = acc;
}

extern "C" void kernel_launch(void* const* d_in, const int* in_sizes, int n_in,
                              void* d_out, int out_size, void* d_ws, size_t ws_size,
                              hipStream_t stream) {
  const float* x   = (const float*)d_in[0];
  const int*   ei  = (const int*)d_in[1];
  const float* W1  = (const float*)d_in[2];
  const float* b1  = (const float*)d_in[3];
  const float* W2  = (const float*)d_in[4];
  const float* b2  = (const float*)d_in[5];
  const float* Wmu = (const float*)d_in[6];
  const float* bmu = (const float*)d_in[7];
  const float* Wlv = (const float*)d_in[8];
  const float* blv = (const float*)d_in[9];
  float* out = (float*)d_out;

  const int N = in_sizes[0] / 32;   // 100000
  const int E = in_sizes[1] / 2;    // 1600000
  const int* src = ei;
  const int* dst = ei + E;

  // workspace layout
  float* A    = (float*)d_ws;                 // N*128  (layer activations)
  float* B    = A + (size_t)N * HID;          // N*128  (gemm output h)
  float* dinv = B + (size_t)N * HID;          // N      (deg -> dinv in place)
  float* g    = dinv + N;                     // 128

  const int T = 256;
  const int gN    = (N + T - 1) / T;
  const int gE    = (E + T - 1) / T;
  const int gF4   = (int)(((size_t)N * (HID / 4) + T - 1) / T);  // N*32 float4 elems
  const int gGemm = N / 16;                                      // 6250
  const int gScat = 2048;                                        // 16384 waves

  // degrees + symmetric normalization
  k_init_deg<<<gN, T, 0, stream>>>(dinv, N);
  k_zero_g<<<1, 128, 0, stream>>>(g);
  k_deg_accum<<<gE, T, 0, stream>>>(dst, dinv, E);
  k_rsqrt_inplace<<<gN, T, 0, stream>>>(dinv, N);

  // layer 1: h = x@W1 ; acc = self + scatter ; A = relu(acc + b1)
  k_gemm_wmma_f32<<<gGemm, T, 0, stream>>>(x, W1, B, N, 32);
  k_selfloop_init<<<gF4, T, 0, stream>>>(B, dinv, A, N);
  k_scatter<<<gScat, T, 0, stream>>>(B, src, dst, dinv, A, E);
  k_bias_relu<<<gF4, T, 0, stream>>>(A, b1, N);

  // layer 2
  k_gemm_wmma_f32<<<gGemm, T, 0, stream>>>(A, W2, B, N, HID);
  k_selfloop_init<<<gF4, T, 0, stream>>>(B, dinv, A, N);
  k_scatter<<<gScat, T, 0, stream>>>(B, src, dst, dinv, A, E);
  k_bias_relu<<<gF4, T, 0, stream>>>(A, b2, N);

  // pool + heads
  k_pool<<<256, 128, 0, stream>>>(A, g, N);
  k_heads<<<1, 128, 0, stream>>>(g, Wmu, bmu, Wlv, blv, out, 1.0f / (float)N);
}